// PFN_30262339567992
// MI455X (gfx1250) — compile-verified
//
#include <hip/hip_runtime.h>
#include <cmath>

// ---- shapes ----
#define LDIM 128
#define BDIM 4
#define DDIM 768
#define HDIM 300
#define G5   1500   // 5*H
#define NT   7
#define NR   12

typedef __attribute__((ext_vector_type(2))) float v2f;
typedef __attribute__((ext_vector_type(8))) float v8f;

__device__ __forceinline__ v8f wmma4(v2f a, v2f b, v8f c) {
  // V_WMMA_F32_16X16X4_F32 : D = A(16x4) * B(4x16) + C(16x16), fp32
  return __builtin_amdgcn_wmma_f32_16x16x4_f32(false, a, false, b, (short)0, c,
                                               false, false);
}

// ---------------------------------------------------------------------------
// Generic C = A @ W^T (+bias).  A: [M x K] row-major (lda), W: [N x ldw] row-
// major with column window starting at woff (K cols used), C: [M x ldc].
// Grid: (M/16, ceil(N/16)), block: 32 (one wave per 16x16 tile). M % 16 == 0,
// K % 4 == 0.  Out-of-range columns read a clamped (valid) W row; the garbage
// lands only in D columns that are never stored.
// ---------------------------------------------------------------------------
__global__ void gemm_wmma(const float* __restrict__ A, int lda,
                          const float* __restrict__ W, int ldw, int woff,
                          const float* __restrict__ bias,
                          float* __restrict__ C, int ldc, int N, int K) {
  const int lane = threadIdx.x & 31;
  const int lm = lane & 15, lh = lane >> 4;
  const int m0 = blockIdx.x << 4, n0 = blockIdx.y << 4;
  const int col = n0 + lm;
  const bool ok = col < N;
  const float* arow = A + (size_t)(m0 + lm) * lda + 2 * lh;
  const float* wrow = W + (size_t)(ok ? col : n0) * ldw + woff + 2 * lh;
  v8f acc = {};
#pragma unroll 4
  for (int k = 0; k < K; k += 4) {
    v2f a, b;
    a.x = arow[k];
    a.y = arow[k + 1];
    b.x = wrow[k];
    b.y = wrow[k + 1];
    acc = wmma4(a, b, acc);
  }
  if (ok) {
    const float bb = bias ? bias[col] : 0.0f;
#pragma unroll
    for (int r = 0; r < 8; ++r)
      C[(size_t)(m0 + r + 8 * lh) * ldc + col] = acc[r] + bb;
  }
}

// ---------------------------------------------------------------------------
// hglob = tanh( cat(A1,A2) @ W^T + bias ).  A1,A2: [512 x 300], W: [300 x 600],
// C: [512 x 300].  Grid (32, 19), block 32.
// ---------------------------------------------------------------------------
__global__ void gemm_cat_tanh(const float* __restrict__ A1,
                              const float* __restrict__ A2,
                              const float* __restrict__ W,
                              const float* __restrict__ bias,
                              float* __restrict__ C) {
  const int lane = threadIdx.x & 31;
  const int lm = lane & 15, lh = lane >> 4;
  const int m0 = blockIdx.x << 4, n0 = blockIdx.y << 4;
  const int col = n0 + lm;
  const bool ok = col < HDIM;
  const float* a1 = A1 + (size_t)(m0 + lm) * HDIM + 2 * lh;
  const float* a2 = A2 + (size_t)(m0 + lm) * HDIM + 2 * lh;
  const float* wrow = W + (size_t)(ok ? col : n0) * (2 * HDIM) + 2 * lh;
  v8f acc = {};
#pragma unroll 2
  for (int k = 0; k < HDIM; k += 4) {   // first half: A1
    v2f a, b;
    a.x = a1[k];
    a.y = a1[k + 1];
    b.x = wrow[k];
    b.y = wrow[k + 1];
    acc = wmma4(a, b, acc);
  }
#pragma unroll 2
  for (int k = 0; k < HDIM; k += 4) {   // second half: A2
    v2f a, b;
    a.x = a2[k];
    a.y = a2[k + 1];
    b.x = wrow[HDIM + k];
    b.y = wrow[HDIM + k + 1];
    acc = wmma4(a, b, acc);
  }
  if (ok) {
    const float bb = bias[col];
#pragma unroll
    for (int r = 0; r < 8; ++r)
      C[(size_t)(m0 + r + 8 * lh) * HDIM + col] = tanhf(acc[r] + bb);
  }
}

// ---------------------------------------------------------------------------
// Persistent recurrent scan over L.  One workgroup, 1024 threads (32 waves).
// State (h, c_in) lives in LDS; both per-step GEMMs use fp32 WMMA.  The M=4
// real rows are padded to 16 by clamped (duplicate) row reads; duplicate D
// rows are discarded at store time.
// ---------------------------------------------------------------------------
__global__ void __launch_bounds__(1024)
scan_kernel(const float* __restrict__ pre,   // [L*B*1500]
            const float* __restrict__ W_hh,  // [1500*300]
            const float* __restrict__ b_hh,  // [1500]
            const float* __restrict__ W_tr,  // [300*900]
            const float* __restrict__ b_tr,  // [300]
            float* __restrict__ h_ner, float* __restrict__ h_re,
            float* __restrict__ h_sh) {      // each [L*B*300]
  __shared__ float sh_h[BDIM * HDIM];      // h state
  __shared__ float sh_cin[BDIM * HDIM];    // carried c
  __shared__ float sh_gates[BDIM * G5];    // gates workspace
  __shared__ float sh_cat[BDIM * 3 * HDIM];// cat(c_re, c_ner, share)

  const int tid = threadIdx.x;
  const int lane = tid & 31, wv = tid >> 5;
  const int lm = lane & 15, lh = lane >> 4;
  const int lmc = lm & 3;                  // clamped A row (duplicates 0..3)

  for (int i = tid; i < BDIM * HDIM; i += 1024) { sh_h[i] = 0.0f; sh_cin[i] = 0.0f; }
  __syncthreads();

  for (int l = 0; l < LDIM; ++l) {
    // ---- Stage A: gates = pre[l] + h @ W_hh^T + b_hh  (M=4->16, N=1500, K=300)
    for (int tn = wv; tn < (G5 + 15) / 16; tn += 32) {
      const int n0 = tn << 4, col = n0 + lm;
      const bool ok = col < G5;
      const float* wrow = W_hh + (size_t)(ok ? col : n0) * HDIM + 2 * lh;
      const float* hrow = sh_h + lmc * HDIM + 2 * lh;
      v8f acc = {};
#pragma unroll 2
      for (int k = 0; k < HDIM; k += 4) {
        v2f a, b;
        a.x = hrow[k];
        a.y = hrow[k + 1];
        b.x = wrow[k];
        b.y = wrow[k + 1];
        acc = wmma4(a, b, acc);
      }
      if (lh == 0 && ok) {
#pragma unroll
        for (int r = 0; r < BDIM; ++r)
          sh_gates[r * G5 + col] =
              acc[r] + pre[(size_t)(l * BDIM + r) * G5 + col] + b_hh[col];
      }
    }
    __syncthreads();

    // ---- Stage B: cummax transforms on 4 gate vectors x 4 batch rows; tanh(c)
    if (wv < 16) {
      const int b = wv >> 2, seg = wv & 3;            // seg 0..3 -> offsets 300..1200
      float* row = sh_gates + b * G5 + HDIM + seg * HDIM;
      const int k0 = lane * 10;
      const int cnt = (k0 < HDIM) ? (HDIM - k0 < 10 ? HDIM - k0 : 10) : 0;
      float x[10];
      float mx = -3.4e38f;
#pragma unroll
      for (int k = 0; k < 10; ++k)
        if (k < cnt) { x[k] = row[k0 + k]; mx = fmaxf(mx, x[k]); }
      for (int d = 16; d; d >>= 1) mx = fmaxf(mx, __shfl_xor(mx, d, 32));
      float s = 0.0f;
#pragma unroll
      for (int k = 0; k < 10; ++k)
        if (k < cnt) { x[k] = expf(x[k] - mx); s += x[k]; }
      float tot = s;
      for (int d = 16; d; d >>= 1) tot += __shfl_xor(tot, d, 32);
      float incl = s;
      for (int d = 1; d < 32; d <<= 1) {
        const float o = __shfl_up(incl, (unsigned)d, 32);
        if (lane >= d) incl += o;
      }
      float run = incl - s;                     // exclusive prefix of lane sums
      const float inv = 1.0f / tot;
      const bool is_eg = (seg == 0) || (seg == 2);  // eg_* -> 1 - cummax
#pragma unroll
      for (int k = 0; k < 10; ++k)
        if (k < cnt) {
          run += x[k];
          const float c = run * inv;
          row[k0 + k] = is_eg ? (1.0f - c) : c;
        }
    } else if (wv < 20) {
      const int b = wv - 16;
      for (int k = lane; k < HDIM; k += 32)
        sh_gates[b * G5 + k] = tanhf(sh_gates[b * G5 + k]);
    }
    __syncthreads();

    // ---- Stage C: combine, emit tanh outputs, build cat matrix
    for (int t = tid; t < BDIM * HDIM; t += 1024) {
      const int b = t / HDIM, hh = t % HDIM;
      const float c   = sh_gates[b * G5 + hh];
      const float egi = sh_gates[b * G5 + HDIM + hh];
      const float rgi = sh_gates[b * G5 + 2 * HDIM + hh];
      const float egc = sh_gates[b * G5 + 3 * HDIM + hh];
      const float rgc = sh_gates[b * G5 + 4 * HDIM + hh];
      const float ovc = rgc * egc, upc = rgc - ovc, dnc = egc - ovc;
      const float ovi = rgi * egi, upi = rgi - ovi, dni = egi - ovi;
      const float ci = sh_cin[b * HDIM + hh];
      const float share = ovi * ci + ovc * c;
      const float cre  = upi * ci + upc * c + share;
      const float cner = dni * ci + dnc * c + share;
      sh_cat[b * 3 * HDIM + hh]            = cre;
      sh_cat[b * 3 * HDIM + HDIM + hh]     = cner;
      sh_cat[b * 3 * HDIM + 2 * HDIM + hh] = share;
      const size_t o = (size_t)(l * BDIM + b) * HDIM + hh;
      h_ner[o] = tanhf(cner);
      h_re[o]  = tanhf(cre);
      h_sh[o]  = tanhf(share);
    }
    __syncthreads();

    // ---- Stage D: c_out = cat @ W_tr^T + b_tr ; h = tanh(c_out), cin = c_out
    if (wv < 19) {
      const int n0 = wv << 4, col = n0 + lm;
      const bool ok = col < HDIM;
      const float* wrow = W_tr + (size_t)(ok ? col : n0) * (3 * HDIM) + 2 * lh;
      const float* crow = sh_cat + lmc * 3 * HDIM + 2 * lh;
      v8f acc = {};
#pragma unroll 2
      for (int k = 0; k < 3 * HDIM; k += 4) {
        v2f a, b;
        a.x = crow[k];
        a.y = crow[k + 1];
        b.x = wrow[k];
        b.y = wrow[k + 1];
        acc = wmma4(a, b, acc);
      }
      if (lh == 0 && ok) {
#pragma unroll
        for (int r = 0; r < BDIM; ++r) {
          const float v = acc[r] + b_tr[col];
          sh_cin[r * HDIM + col] = v;
          sh_h[r * HDIM + col] = tanhf(v);
        }
      }
    }
    __syncthreads();
  }
}

// ---------------------------------------------------------------------------
// g[b,h] = max_l hglob[l,b,h]
// ---------------------------------------------------------------------------
__global__ void colmax_kernel(const float* __restrict__ hg, float* __restrict__ g) {
  const int t = blockIdx.x * blockDim.x + threadIdx.x;
  if (t >= BDIM * HDIM) return;
  const int b = t / HDIM, h = t % HDIM;
  float m = -3.4e38f;
  for (int l = 0; l < LDIM; ++l)
    m = fmaxf(m, hg[(size_t)(l * BDIM + b) * HDIM + h]);
  g[t] = m;
}

// ---------------------------------------------------------------------------
// R[b,n] = sum_k g[b,k] * hW[n, 600+k] + hb[n]
// ---------------------------------------------------------------------------
__global__ void global_proj_kernel(const float* __restrict__ g,
                                   const float* __restrict__ hW,
                                   const float* __restrict__ hb,
                                   float* __restrict__ R) {
  const int t = blockIdx.x * blockDim.x + threadIdx.x;
  if (t >= BDIM * HDIM) return;
  const int b = t / HDIM, n = t % HDIM;
  const float* wr = hW + (size_t)n * (3 * HDIM) + 2 * HDIM;
  const float* gr = g + b * HDIM;
  float s = hb[n];
  for (int k = 0; k < HDIM; ++k) s += gr[k] * wr[k];
  R[t] = s;
}

// ---------------------------------------------------------------------------
// Pair kernel: one wave per (i, j, b).
// v = P[i,b,:] + Q[j,b,:] + R[b,:] ; LN -> ELU -> sigmoid(v @ tW^T + tb) * mask
// ---------------------------------------------------------------------------
template <int T>
__global__ void __launch_bounds__(256)
pair_kernel(const float* __restrict__ P, const float* __restrict__ Q,
            const float* __restrict__ R, const float* __restrict__ mask,
            const float* __restrict__ lng, const float* __restrict__ lnb,
            const float* __restrict__ tW, const float* __restrict__ tb,
            float* __restrict__ out, int diag) {
  __shared__ float s_g[HDIM], s_b[HDIM], s_tw[T * HDIM], s_tb[T],
      s_R[BDIM * HDIM];
  const int tid = threadIdx.x;
  for (int i = tid; i < HDIM; i += 256) { s_g[i] = lng[i]; s_b[i] = lnb[i]; }
  for (int i = tid; i < T * HDIM; i += 256) s_tw[i] = tW[i];
  for (int i = tid; i < BDIM * HDIM; i += 256) s_R[i] = R[i];
  if (tid < T) s_tb[tid] = tb[tid];
  __syncthreads();

  const int lane = tid & 31, wv = tid >> 5;
  const int pid = blockIdx.x * 8 + wv;         // 0 .. 65535
  const int b = pid & 3, j = (pid >> 2) & (LDIM - 1), i = pid >> 9;

  const int k0 = lane * 10;
  const int cnt = (k0 < HDIM) ? (HDIM - k0 < 10 ? HDIM - k0 : 10) : 0;
  const float* Pp = P + (size_t)(i * BDIM + b) * HDIM + k0;
  const float* Qp = Q + (size_t)(j * BDIM + b) * HDIM + k0;
  const float* Rp = s_R + b * HDIM + k0;

  float v[10];
  float s = 0.0f, sq = 0.0f;
#pragma unroll
  for (int k = 0; k < 10; ++k)
    if (k < cnt) {
      const float x = Pp[k] + Qp[k] + Rp[k];
      v[k] = x; s += x; sq += x * x;
    }
  for (int d = 16; d; d >>= 1) { s += __shfl_xor(s, d, 32); sq += __shfl_xor(sq, d, 32); }
  const float mu = s * (1.0f / HDIM);
  const float var = sq * (1.0f / HDIM) - mu * mu;
  const float rs = rsqrtf(var + 1e-5f);

  float acc[T];
#pragma unroll
  for (int t = 0; t < T; ++t) acc[t] = 0.0f;
#pragma unroll
  for (int k = 0; k < 10; ++k)
    if (k < cnt) {
      const float y = (v[k] - mu) * rs * s_g[k0 + k] + s_b[k0 + k];
      const float e = (y > 0.0f) ? y : (expf(y) - 1.0f);
#pragma unroll
      for (int t = 0; t < T; ++t) acc[t] += e * s_tw[t * HDIM + k0 + k];
    }
#pragma unroll
  for (int t = 0; t < T; ++t)
    for (int d = 16; d; d >>= 1) acc[t] += __shfl_xor(acc[t], d, 32);

  float m = mask[i * BDIM + b] * mask[j * BDIM + b];
  if (diag && j < i) m = 0.0f;
#pragma unroll
  for (int t = 0; t < T; ++t)
    if (lane == t)
      out[(size_t)pid * T + t] = m / (1.0f + expf(-(acc[t] + s_tb[t])));
}

// ---------------------------------------------------------------------------
extern "C" void kernel_launch(void* const* d_in, const int* in_sizes, int n_in,
                              void* d_out, int out_size, void* d_ws,
                              size_t ws_size, hipStream_t stream) {
  const float* x      = (const float*)d_in[0];
  const float* mask   = (const float*)d_in[1];
  const float* W_ih   = (const float*)d_in[2];
  const float* b_ih   = (const float*)d_in[3];
  const float* W_hh   = (const float*)d_in[4];
  const float* b_hh   = (const float*)d_in[5];
  const float* W_tr   = (const float*)d_in[6];
  const float* b_tr   = (const float*)d_in[7];
  const float* n_W    = (const float*)d_in[8];
  const float* n_b    = (const float*)d_in[9];
  const float* ner_hW = (const float*)d_in[10];
  const float* ner_hb = (const float*)d_in[11];
  const float* ner_g  = (const float*)d_in[12];
  const float* ner_be = (const float*)d_in[13];
  const float* ner_tW = (const float*)d_in[14];
  const float* ner_tb = (const float*)d_in[15];
  const float* r_W    = (const float*)d_in[16];
  const float* r_b    = (const float*)d_in[17];
  const float* re_hW  = (const float*)d_in[18];
  const float* re_hb  = (const float*)d_in[19];
  const float* re_g   = (const float*)d_in[20];
  const float* re_be  = (const float*)d_in[21];
  const float* re_tW  = (const float*)d_in[22];
  const float* re_tb  = (const float*)d_in[23];

  float* ws = (float*)d_ws;
  float* pre   = ws;                         // 512*1500 = 768000 floats
  float* h_ner = ws + 768000;                // 512*300
  float* h_re  = h_ner + 512 * HDIM;
  float* h_sh  = h_re + 512 * HDIM;
  float* gbuf  = h_sh + 512 * HDIM;          // 1200
  float* Rbuf  = gbuf + 1280;                // 1200
  // reuse the `pre` region after the scan:
  float* hglob = ws;                         // 512*300
  float* P     = ws + 512 * HDIM;            // 512*300
  float* Q     = ws + 2 * 512 * HDIM;        // 512*300

  float* out_ner = (float*)d_out;                              // [L,L,B,NT]
  float* out_re  = out_ner + (size_t)LDIM * LDIM * BDIM * NT;  // [L,L,B,NR]

  const int MT = 512 / 16;  // M tiles for [L*B, *] GEMMs

  // 1) pre = x @ W_ih^T + b_ih
  gemm_wmma<<<dim3(MT, (G5 + 15) / 16), 32, 0, stream>>>(
      x, DDIM, W_ih, DDIM, 0, b_ih, pre, G5, G5, DDIM);

  // 2) recurrent scan
  scan_kernel<<<1, 1024, 0, stream>>>(pre, W_hh, b_hh, W_tr, b_tr, h_ner, h_re,
                                      h_sh);

  // 3) NER head
  gemm_cat_tanh<<<dim3(MT, (HDIM + 15) / 16), 32, 0, stream>>>(h_sh, h_ner, n_W,
                                                               n_b, hglob);
  colmax_kernel<<<5, 256, 0, stream>>>(hglob, gbuf);
  global_proj_kernel<<<5, 256, 0, stream>>>(gbuf, ner_hW, ner_hb, Rbuf);
  gemm_wmma<<<dim3(MT, (HDIM + 15) / 16), 32, 0, stream>>>(
      h_ner, HDIM, ner_hW, 3 * HDIM, 0, (const float*)nullptr, P, HDIM, HDIM, HDIM);
  gemm_wmma<<<dim3(MT, (HDIM + 15) / 16), 32, 0, stream>>>(
      h_ner, HDIM, ner_hW, 3 * HDIM, HDIM, (const float*)nullptr, Q, HDIM, HDIM, HDIM);
  pair_kernel<NT><<<LDIM * LDIM * BDIM / 8, 256, 0, stream>>>(
      P, Q, Rbuf, mask, ner_g, ner_be, ner_tW, ner_tb, out_ner, 1);

  // 4) RE head
  gemm_cat_tanh<<<dim3(MT, (HDIM + 15) / 16), 32, 0, stream>>>(h_sh, h_re, r_W,
                                                               r_b, hglob);
  colmax_kernel<<<5, 256, 0, stream>>>(hglob, gbuf);
  global_proj_kernel<<<5, 256, 0, stream>>>(gbuf, re_hW, re_hb, Rbuf);
  gemm_wmma<<<dim3(MT, (HDIM + 15) / 16), 32, 0, stream>>>(
      h_re, HDIM, re_hW, 3 * HDIM, 0, (const float*)nullptr, P, HDIM, HDIM, HDIM);
  gemm_wmma<<<dim3(MT, (HDIM + 15) / 16), 32, 0, stream>>>(
      h_re, HDIM, re_hW, 3 * HDIM, HDIM, (const float*)nullptr, Q, HDIM, HDIM, HDIM);
  pair_kernel<NR><<<LDIM * LDIM * BDIM / 8, 256, 0, stream>>>(
      P, Q, Rbuf, mask, re_g, re_be, re_tW, re_tb, out_re, 0);
}